// HANLayer_18545668784544
// MI455X (gfx1250) — compile-verified
//
#include <hip/hip_runtime.h>
#include <math.h>

typedef float v2f __attribute__((ext_vector_type(2)));
typedef float v8f __attribute__((ext_vector_type(8)));

#define IN_F  128
#define OUT_F 64

// ---------------------------------------------------------------------------
// Kernel 1: fused dual GEMM  feat = h @ W_gat,  support = h @ W_gc
// 256 threads = 8 waves; each wave owns 16 rows of h; weights staged in LDS.
// WMMA f32 16x16x4: A 16x4 (2 VGPR/lane), B 4x16 (2 VGPR/lane), C/D 16x16 (8 VGPR).
// Layout per ISA 7.12.2: lanes 0-15 hold K=0,1 ; lanes 16-31 hold K=2,3.
// ---------------------------------------------------------------------------
__global__ __launch_bounds__(256)
void gemm_fused_kernel(const float* __restrict__ h,
                       const float* __restrict__ Wgat,   // [128][64]
                       const float* __restrict__ Wgc,    // [128][64]
                       float* __restrict__ feat,
                       float* __restrict__ support,
                       int n)
{
    __shared__ float wlds[2 * IN_F * OUT_F];   // 64 KB (320 KB/WGP available)

    const int tid = threadIdx.x;
    for (int i = tid; i < IN_F * OUT_F; i += 256) {
        wlds[i]                 = Wgat[i];
        wlds[IN_F * OUT_F + i]  = Wgc[i];
    }
    __syncthreads();

    const int wave = tid >> 5;          // 0..7
    const int lane = tid & 31;
    const int ln   = lane & 15;         // 0..15
    const int hi   = lane >> 4;         // 0 or 1 (K offset +2)

    const int rowbase = (blockIdx.x * 8 + wave) * 16;
    int arow = rowbase + ln;
    if (arow >= n) arow = n - 1;        // clamp: keep EXEC all-ones for WMMA
    const float* __restrict__ hrow = h + (size_t)arow * IN_F;

    v8f acc[8] = {};                    // tiles 0..3: W_gat n-tiles, 4..7: W_gc

    for (int kc = 0; kc < IN_F; kc += 4) {
        const int k0 = kc + 2 * hi;     // this lane's K rows: k0, k0+1
        v2f a;
        a.x = hrow[k0];
        a.y = hrow[k0 + 1];
#pragma unroll
        for (int t = 0; t < 8; ++t) {
            const int mtx = t >> 2;     // 0 = gat, 1 = gc
            const int nt  = t & 3;      // n-tile (16 cols each)
            const float* wl = wlds + mtx * (IN_F * OUT_F) + k0 * OUT_F + nt * 16 + ln;
            v2f b;
            b.x = wl[0];                // row k0
            b.y = wl[OUT_F];            // row k0+1
            acc[t] = __builtin_amdgcn_wmma_f32_16x16x4_f32(
                false, a, false, b, (short)0, acc[t], false, false);
        }
    }

    // D layout: lane l, vgpr v -> element (M = v + 8*(l>=16), Ncol = nt*16 + l%16)
    if (rowbase + 16 <= n) {
        // fast path (wave-uniform): all 16 rows valid -> unguarded coalesced stores
#pragma unroll
        for (int t = 0; t < 8; ++t) {
            const int mtx = t >> 2;
            const int nt  = t & 3;
            float* outp = (mtx == 0) ? feat : support;
#pragma unroll
            for (int v = 0; v < 8; ++v) {
                const int row = rowbase + v + 8 * hi;
                outp[(size_t)row * OUT_F + nt * 16 + ln] = acc[t][v];
            }
        }
    } else {
        // slow path: only the final partial wave takes this
#pragma unroll
        for (int t = 0; t < 8; ++t) {
            const int mtx = t >> 2;
            const int nt  = t & 3;
            float* outp = (mtx == 0) ? feat : support;
#pragma unroll
            for (int v = 0; v < 8; ++v) {
                const int row = rowbase + v + 8 * hi;
                if (row < n)
                    outp[(size_t)row * OUT_F + nt * 16 + ln] = acc[t][v];
            }
        }
    }
}

// ---------------------------------------------------------------------------
// Kernel 2: el = feat @ attn_l, er = feat @ attn_r   (warp per row)
// ---------------------------------------------------------------------------
__global__ __launch_bounds__(256)
void eler_kernel(const float* __restrict__ feat,
                 const float* __restrict__ attn_l,
                 const float* __restrict__ attn_r,
                 float* __restrict__ el, float* __restrict__ er, int n)
{
    const int warp = (int)((blockIdx.x * blockDim.x + threadIdx.x) >> 5);
    const int lane = threadIdx.x & 31;
    if (warp >= n) return;
    const float* f = feat + (size_t)warp * OUT_F;
    float l = f[lane] * attn_l[lane] + f[lane + 32] * attn_l[lane + 32];
    float r = f[lane] * attn_r[lane] + f[lane + 32] * attn_r[lane + 32];
#pragma unroll
    for (int m = 16; m; m >>= 1) {
        l += __shfl_xor(l, m, 32);
        r += __shfl_xor(r, m, 32);
    }
    if (lane == 0) { el[warp] = l; er[warp] = r; }
}

// ---------------------------------------------------------------------------
// Kernel 3: init accumulators. ws layout: [m(n) | s(n) | agg(n*64) | gc(n*64)]
// m <- -3e38 (segment-max identity), rest <- 0
// ---------------------------------------------------------------------------
__global__ void init_kernel(float* __restrict__ ws, int n)
{
    const int total = n * 130;
    for (int i = blockIdx.x * blockDim.x + threadIdx.x; i < total;
         i += gridDim.x * blockDim.x)
        ws[i] = (i < n) ? -3.0e38f : 0.0f;
}

// ---------------------------------------------------------------------------
// Kernel 4: e = leaky_relu(el[src]+er[dst]); m[dst] = max(m[dst], e)
// ---------------------------------------------------------------------------
__global__ void edge_max_kernel(const float* __restrict__ el,
                                const float* __restrict__ er,
                                const int* __restrict__ src,
                                const int* __restrict__ dst,
                                float* __restrict__ ebuf,
                                float* __restrict__ m, int e)
{
    const int i = blockIdx.x * blockDim.x + threadIdx.x;
    if (i >= e) return;
    float x = el[src[i]] + er[dst[i]];
    x = (x > 0.0f) ? x : 0.2f * x;
    ebuf[i] = x;
    atomicMax(&m[dst[i]], x);
}

// ---------------------------------------------------------------------------
// Kernel 5: ex = exp(e - m[dst]); s[dst] += ex; ebuf[i] = ex
// ---------------------------------------------------------------------------
__global__ void edge_expsum_kernel(const int* __restrict__ dst,
                                   const float* __restrict__ m,
                                   float* __restrict__ ebuf,
                                   float* __restrict__ s, int e)
{
    const int i = blockIdx.x * blockDim.x + threadIdx.x;
    if (i >= e) return;
    const int d = dst[i];
    const float ex = __expf(ebuf[i] - m[d]);
    ebuf[i] = ex;
    atomicAdd(&s[d], ex);
}

// ---------------------------------------------------------------------------
// Kernel 6 (used twice): out[gdst] += w[i] * nodefeat[gsrc]   (warp per edge)
// ---------------------------------------------------------------------------
__global__ __launch_bounds__(256)
void edge_aggregate_kernel(const float* __restrict__ w,
                           const int* __restrict__ gsrc,
                           const int* __restrict__ gdst,
                           const float* __restrict__ nodefeat,
                           float* __restrict__ out, int e)
{
    const int edge = (int)((blockIdx.x * blockDim.x + threadIdx.x) >> 5);
    const int lane = threadIdx.x & 31;
    if (edge >= e) return;
    const float a = w[edge];
    const float* f = nodefeat + (size_t)gsrc[edge] * OUT_F;
    float* o = out + (size_t)gdst[edge] * OUT_F;
    atomicAdd(&o[lane],      a * f[lane]);
    atomicAdd(&o[lane + 32], a * f[lane + 32]);
}

// ---------------------------------------------------------------------------
// Kernel 7: out[n][0:64]  = elu(agg/s + bias_gat)
//           out[n][64:128]= elu(gc + bias_gc)
// ---------------------------------------------------------------------------
__global__ void finalize_kernel(const float* __restrict__ agg,
                                const float* __restrict__ s,
                                const float* __restrict__ gc,
                                const float* __restrict__ bias_gat,
                                const float* __restrict__ bias_gc,
                                float* __restrict__ out, int n)
{
    const int idx = blockIdx.x * blockDim.x + threadIdx.x;
    if (idx >= n * 128) return;
    const int nn = idx >> 7;
    const int c  = idx & 127;
    float v;
    if (c < 64) {
        const float sv  = s[nn];
        const float inv = (sv > 0.0f) ? (1.0f / sv) : 0.0f;
        v = agg[(size_t)nn * OUT_F + c] * inv + bias_gat[c];
    } else {
        v = gc[(size_t)nn * OUT_F + (c - 64)] + bias_gc[c - 64];
    }
    v = (v > 0.0f) ? v : expm1f(v);   // elu
    out[idx] = v;
}

// ---------------------------------------------------------------------------
extern "C" void kernel_launch(void* const* d_in, const int* in_sizes, int n_in,
                              void* d_out, int out_size, void* d_ws, size_t ws_size,
                              hipStream_t stream)
{
    const float* h        = (const float*)d_in[0];
    const int*   src0     = (const int*)  d_in[1];
    const int*   dst0     = (const int*)  d_in[2];
    const int*   row2     = (const int*)  d_in[3];
    const int*   col2     = (const int*)  d_in[4];
    const float* adj_val  = (const float*)d_in[5];
    const float* W_gat    = (const float*)d_in[6];
    const float* attn_l   = (const float*)d_in[7];
    const float* attn_r   = (const float*)d_in[8];
    const float* bias_gat = (const float*)d_in[9];
    const float* W_gc     = (const float*)d_in[10];
    const float* bias_gc  = (const float*)d_in[11];

    const int n = in_sizes[0] / IN_F;   // 50000
    const int e = in_sizes[1];          // 800000

    // workspace layout (floats)
    float* ws      = (float*)d_ws;
    float* m       = ws;                       // n
    float* s       = ws + (size_t)n;           // n
    float* agg     = ws + (size_t)2 * n;       // n*64
    float* gc      = ws + (size_t)66 * n;      // n*64
    float* feat    = ws + (size_t)130 * n;     // n*64
    float* support = ws + (size_t)194 * n;     // n*64
    float* el      = ws + (size_t)258 * n;     // n
    float* er      = ws + (size_t)259 * n;     // n
    float* ebuf    = ws + (size_t)260 * n;     // e

    float* out = (float*)d_out;

    // 1) dual GEMM (WMMA f32)
    gemm_fused_kernel<<<(n + 127) / 128, 256, 0, stream>>>(
        h, W_gat, W_gc, feat, support, n);

    // 2) el / er
    eler_kernel<<<(n + 7) / 8, 256, 0, stream>>>(feat, attn_l, attn_r, el, er, n);

    // 3) init m / s / agg / gc
    init_kernel<<<2048, 256, 0, stream>>>(ws, n);

    // 4) leaky-relu + segment max
    edge_max_kernel<<<(e + 255) / 256, 256, 0, stream>>>(el, er, src0, dst0, ebuf, m, e);

    // 5) exp + segment sum
    edge_expsum_kernel<<<(e + 255) / 256, 256, 0, stream>>>(dst0, m, ebuf, s, e);

    // 6) GAT aggregation: agg[dst] += ex * feat[src]
    edge_aggregate_kernel<<<(e + 7) / 8, 256, 0, stream>>>(ebuf, src0, dst0, feat, agg, e);

    // 7) GC aggregation: gc[row2] += adj_val * support[col2]
    edge_aggregate_kernel<<<(e + 7) / 8, 256, 0, stream>>>(adj_val, col2, row2, support, gc, e);

    // 8) finalize + concat
    finalize_kernel<<<(n * 128 + 255) / 256, 256, 0, stream>>>(
        agg, s, gc, bias_gat, bias_gc, out, n);
}